// PEGH_63771674411470
// MI455X (gfx1250) — compile-verified
//
#include <hip/hip_runtime.h>
#include <stdint.h>

#define HH   256
#define WW   256
#define NPTS 32768
#define DIMC 512
#define PAD  3
#define PW   (WW + 2*PAD)          // 262
#define PH   (HH + 2*PAD)          // 262
#define TILE 16
#define HALO (TILE + 2*PAD)        // 22
#define CB   64                    // channels per block
#define NCB  (DIMC / CB)           // 8
#define DENSE_FLOATS ((size_t)PH * PW * DIMC)   // 35,145,728 floats (~140.6 MB)

typedef __attribute__((ext_vector_type(2))) float        f32x2;
typedef __attribute__((ext_vector_type(4))) float        f32x4;
typedef __attribute__((ext_vector_type(4))) unsigned int u32x4;
typedef __attribute__((ext_vector_type(8))) int          i32x8;
typedef __attribute__((ext_vector_type(4))) int          i32x4;

// ---------------- cell -> point map ----------------
__global__ void map_init_kernel(int* __restrict__ map) {
    map[blockIdx.x * blockDim.x + threadIdx.x] = -1;
}

__global__ void map_scatter_kernel(const int* __restrict__ pos, int* __restrict__ map) {
    int n = blockIdx.x * blockDim.x + threadIdx.x;
    if (n < NPTS) {
        int h = pos[2 * n + 0];
        int w = pos[2 * n + 1];
        map[h * WW + w] = n;
    }
}

// ---------------- combined 7x7 per-channel weights + bias ----------------
// T[k][c], k = dj*7 + di; tap (di,dj) multiplies dense[h + dj - 3][w + di - 3][c].
__global__ void build_weights_kernel(const float* __restrict__ w1, const float* __restrict__ b1,
                                     const float* __restrict__ w2, const float* __restrict__ b2,
                                     const float* __restrict__ w3, const float* __restrict__ b3,
                                     float* __restrict__ T, float* __restrict__ Bsum) {
    int gid = blockIdx.x * blockDim.x + threadIdx.x;   // 49*512 threads
    int k  = gid / DIMC;
    int c  = gid % DIMC;
    int dj = k / 7;
    int di = k % 7;
    float v = w3[c * 49 + di * 7 + dj];
    if (di >= 1 && di <= 5 && dj >= 1 && dj <= 5)
        v += w2[c * 25 + (di - 1) * 5 + (dj - 1)];
    if (di >= 2 && di <= 4 && dj >= 2 && dj <= 4)
        v += w1[c * 9 + (di - 2) * 3 + (dj - 2)];
    T[k * DIMC + c] = v;
    if (gid < DIMC) Bsum[gid] = b1[gid] + b2[gid] + b3[gid];
}

// ---------------- single-pass fill of padded dense grid ----------------
__global__ void fill_dense_kernel(const float* __restrict__ x, const int* __restrict__ map,
                                  float* __restrict__ dense) {
    const int cell = blockIdx.x;            // 0 .. PH*PW-1
    const int t = threadIdx.x;              // 128 threads x float4 = 512 floats
    const int ph = cell / PW;
    const int pw = cell - ph * PW;
    f32x4* dst = (f32x4*)(dense + (size_t)cell * DIMC);

    int n = -1;
    if (ph >= PAD && ph < PAD + HH && pw >= PAD && pw < PAD + WW)
        n = map[(ph - PAD) * WW + (pw - PAD)];

    if (n >= 0) {
        const f32x4* src = (const f32x4*)(x + (size_t)n * DIMC);
        dst[t] = __builtin_nontemporal_load(src + t);   // x read once: keep L2 for dense
    } else {
        f32x4 z = {0.f, 0.f, 0.f, 0.f};
        dst[t] = z;
    }
}

// ---------------- LDS-tiled combined 7x7 depthwise stencil ----------------
// Grid: 16x16 spatial tiles x 8 channel blocks = 2048 blocks, 256 threads.
// Halo tile (22x22 cells x 64 ch = 124KB) moved by ONE Tensor Data Mover
// descriptor per workgroup (3D tile: 64 contiguous ch, 22 cols, 22 rows).
__global__ void __launch_bounds__(256, 1)
stencil_kernel(const float* __restrict__ dense, const float* __restrict__ T,
               const float* __restrict__ Bsum, const int* __restrict__ map,
               float* __restrict__ out) {
    extern __shared__ float lds[];
    const int t  = threadIdx.x;
    const int bid = blockIdx.x;
    const int tx = bid & 15;            // tile col
    const int ty = (bid >> 4) & 15;     // tile row
    const int cb = bid >> 8;            // channel block 0..7
    const int c0 = cb * CB;

    const uint32_t lds_base = (uint32_t)(uintptr_t)&lds[0];   // low 32 bits == LDS offset

    // ---- wave 0 issues the TDM descriptor (TENSORcnt path, CDNA5) ----
    if (t < 32) {
        const uint64_t ga = (uint64_t)(uintptr_t)
            (dense + ((size_t)(ty * TILE) * PW + (size_t)(tx * TILE)) * DIMC + c0);
        u32x4 g0;
        g0.x = 1u;                                    // count=1: valid user descriptor
        g0.y = lds_base;                              // lds_addr (bytes)
        g0.z = (uint32_t)ga;                          // global_addr[31:0]
        g0.w = (uint32_t)(ga >> 32) | (2u << 30);     // global_addr[56:32] | type=2
        i32x8 g1;
        g1[0] = 0x00020000;                           // workgroup_mask=0, data_size=4B
        g1[1] = CB << 16;                             // tensor_dim0 = 64
        g1[2] = HALO << 16;                           // tensor_dim1 = 22
        g1[3] = CB << 16;                             // tile_dim0 = 64
        g1[4] = HALO | (HALO << 16);                  // tile_dim1 = 22, tile_dim2 = 22
        g1[5] = DIMC;                                 // tensor_dim0_stride = 512 elems
        g1[6] = (int)(((PW * DIMC) & 0xFFFF) << 16);  // dim1_stride[15:0] = 0x0C00
        g1[7] = (PW * DIMC) >> 16;                    // dim1_stride[47:16] = 0x2
        i32x4 g2 = {HALO, 0, 0, 0};                   // tensor_dim2 = 22
        i32x4 g3 = {0, 0, 0, 0};                      // dims 3/4 unused
        i32x8 g4 = {0, 0, 0, 0, 0, 0, 0, 0};          // extra group (6-arg builtin): unused
        __builtin_amdgcn_tensor_load_to_lds(g0, g1, g2, g3, g4, 0);
    }

    // ---- preload per-lane weights (overlaps with the TDM DMA) ----
    const int lane = t & 31;
    const int wv   = t >> 5;            // wave 0..7
    f32x2 wt[49];
#pragma unroll
    for (int k = 0; k < 49; ++k)
        wt[k] = *(const f32x2*)(T + (size_t)k * DIMC + c0 + lane * 2);
    const f32x2 bias = *(const f32x2*)(Bsum + c0 + lane * 2);

    if (t < 32) __builtin_amdgcn_s_wait_tensorcnt(0);  // only issuing wave waits
    __syncthreads();

    // ---- per-cell 49-tap stencil; wave <-> cell, lane <-> channel pair ----
    for (int cellL = wv; cellL < TILE * TILE; cellL += 8) {
        int chh = cellL >> 4;
        int cww = cellL & 15;
        int n = map[(ty * TILE + chh) * WW + (tx * TILE + cww)];  // wave-uniform
        if (n < 0) continue;                                      // uniform skip
        f32x2 acc = bias;
#pragma unroll
        for (int j = 0; j < 7; ++j) {
#pragma unroll
            for (int i = 0; i < 7; ++i) {
                const f32x2 v = *(const f32x2*)
                    &lds[(((chh + j) * HALO) + (cww + i)) * CB + lane * 2];
                const f32x2 w = wt[j * 7 + i];
                acc.x = fmaf(w.x, v.x, acc.x);
                acc.y = fmaf(w.y, v.y, acc.y);
            }
        }
        // + x_dense at the occupied cell (== center value, CSEs with tap (3,3))
        const f32x2 ctr = *(const f32x2*)
            &lds[(((chh + 3) * HALO) + (cww + 3)) * CB + lane * 2];
        acc.x += ctr.x;
        acc.y += ctr.y;
        // out written once, never re-read: NT store keeps dense hot in L2
        __builtin_nontemporal_store(acc, (f32x2*)(out + (size_t)n * DIMC + c0 + lane * 2));
    }
}

extern "C" void kernel_launch(void* const* d_in, const int* in_sizes, int n_in,
                              void* d_out, int out_size, void* d_ws, size_t ws_size,
                              hipStream_t stream) {
    const float* x   = (const float*)d_in[0];
    const int*   pos = (const int*)  d_in[1];
    const float* w1  = (const float*)d_in[2];
    const float* b1  = (const float*)d_in[3];
    const float* w2  = (const float*)d_in[4];
    const float* b2  = (const float*)d_in[5];
    const float* w3  = (const float*)d_in[6];
    const float* b3  = (const float*)d_in[7];
    float* out = (float*)d_out;

    // workspace layout
    float* dense = (float*)d_ws;                       // 35,145,728 floats (~140.6 MB)
    float* T     = dense + DENSE_FLOATS;               // 49*512 floats
    float* Bsum  = T + 49 * DIMC;                      // 512 floats
    int*   map   = (int*)(Bsum + DIMC);                // 65536 ints

    map_init_kernel<<<(HH * WW) / 256, 256, 0, stream>>>(map);
    map_scatter_kernel<<<NPTS / 256, 256, 0, stream>>>(pos, map);
    build_weights_kernel<<<(49 * DIMC) / 256, 256, 0, stream>>>(w1, b1, w2, b2, w3, b3, T, Bsum);
    fill_dense_kernel<<<PH * PW, 128, 0, stream>>>(x, map, dense);

    const int stencil_lds = HALO * HALO * CB * (int)sizeof(float);  // 123,904 B
    stencil_kernel<<<16 * 16 * NCB, 256, stencil_lds, stream>>>(dense, T, Bsum, map, out);
}